// GraphNet_38448547234818
// MI455X (gfx1250) — compile-verified
//
#include <hip/hip_runtime.h>

typedef float v2f __attribute__((ext_vector_type(2)));
typedef float v8f __attribute__((ext_vector_type(8)));

#define BN_EPS 1e-5f

// Workspace layout (floats):
//   [  0..191] Bw1 fragments : v2f [3][32]   (BN-folded W1, padded 12->16, WMMA B lane layout)
//   [192..383] Bw2 fragments : v2f [3][32]   (W2, padded,             WMMA B lane layout)
//   [384..399] b1x[16]  (BN-folded bias1, padded)
//   [400..415] b2x[16]  (bias2, padded)
//
// WMMA f32 16x16x4 lane layout assumption (mirrors documented A layout):
//   A/B fragment j (K = 4j..4j+3):  lanes 0-15 hold K = {4j, 4j+1},
//                                   lanes 16-31 hold K = {4j+2, 4j+3}.

__global__ void prep_kernel(const float* __restrict__ bn_scale,
                            const float* __restrict__ bn_bias,
                            const float* __restrict__ bn_mean,
                            const float* __restrict__ bn_var,
                            const float* __restrict__ W1,
                            const float* __restrict__ b1,
                            const float* __restrict__ W2,
                            const float* __restrict__ b2,
                            float* __restrict__ ws) {
  int tid = threadIdx.x;
  if (tid < 96) {
    // B-matrix fragments for W1' (BN-folded) and W2
    int j    = tid >> 5;
    int lane = tid & 31;
    int hi   = lane >> 4;
    int n    = lane & 15;
    int k0   = 4 * j + 2 * hi;
    float v1[2], v2[2];
#pragma unroll
    for (int u = 0; u < 2; ++u) {
      int k = k0 + u;
      float a = 0.f, b = 0.f;
      if (k < 12 && n < 12) {
        float s = bn_scale[k] * rsqrtf(bn_var[k] + BN_EPS);
        a = s * W1[k * 12 + n];
        b = W2[k * 12 + n];
      }
      v1[u] = a;
      v2[u] = b;
    }
    ws[(j * 32 + lane) * 2 + 0]       = v1[0];
    ws[(j * 32 + lane) * 2 + 1]       = v1[1];
    ws[192 + (j * 32 + lane) * 2 + 0] = v2[0];
    ws[192 + (j * 32 + lane) * 2 + 1] = v2[1];
  } else if (tid < 112) {
    // b1' = b1 + t @ W1,  t[k] = bias[k] - mean[k]*s[k]
    int n = tid - 96;
    float acc = 0.f;
    if (n < 12) {
      acc = b1[n];
#pragma unroll
      for (int k = 0; k < 12; ++k) {
        float s = bn_scale[k] * rsqrtf(bn_var[k] + BN_EPS);
        float t = bn_bias[k] - bn_mean[k] * s;
        acc += t * W1[k * 12 + n];
      }
    }
    ws[384 + n] = acc;
  } else if (tid < 128) {
    int n = tid - 112;
    ws[400 + n] = (n < 12) ? b2[n] : 0.f;
  }
}

__global__ void zero_kernel(float* __restrict__ out, int nelem) {
  int i = blockIdx.x * blockDim.x + threadIdx.x;
  if (i < nelem) out[i] = 0.f;
}

__global__ __launch_bounds__(256) void edge_wmma_kernel(
    const float* __restrict__ edges, const int* __restrict__ receivers,
    const float* __restrict__ ws, float* __restrict__ out, int ntiles) {
  __shared__ float hbuf[8][256];  // 1 KB per wave, wave-private

  int lane = threadIdx.x & 31;
  int wave = threadIdx.x >> 5;
  int hi   = lane >> 4;
  int n    = lane & 15;

  // Loop-invariant constant fragments: held in registers for the whole
  // grid-stride loop (one load per wave, not per tile).
  const v2f* B = (const v2f*)ws;
  v2f bw1_0 = B[0 * 32 + lane];
  v2f bw1_1 = B[1 * 32 + lane];
  v2f bw1_2 = B[2 * 32 + lane];
  v2f bw2_0 = B[96 + 0 * 32 + lane];
  v2f bw2_1 = B[96 + 1 * 32 + lane];
  v2f bw2_2 = B[96 + 2 * 32 + lane];
  float c1 = ws[384 + n];
  float c2 = ws[400 + n];

  float* hb = &hbuf[wave][0];
  int wave_stride = gridDim.x * 8;  // waves in grid
  long row_stride = (long)wave_stride * 16 * 12;  // floats between a wave's tiles

  for (int tile = blockIdx.x * 8 + wave; tile < ntiles; tile += wave_stride) {
    int e0 = tile * 16;

    // A fragments: lane's matrix row is M = lane&15, K chunk selected by hi
    const float* erow = edges + (size_t)(e0 + n) * 12;
    v2f a0 = *(const v2f*)(erow + 0 + 2 * hi);
    v2f a1 = *(const v2f*)(erow + 4 + 2 * hi);
    v2f a2 = *(const v2f*)(erow + 8 + 2 * hi);

    // Prefetch next iteration's edge rows (global_prefetch_b8)
    __builtin_prefetch(erow + row_stride, 0, 3);

    // GEMM1: h = edges @ W1' + b1'
    v8f acc = {c1, c1, c1, c1, c1, c1, c1, c1};
    acc = __builtin_amdgcn_wmma_f32_16x16x4_f32(false, a0, false, bw1_0, (short)0, acc, false, false);
    acc = __builtin_amdgcn_wmma_f32_16x16x4_f32(false, a1, false, bw1_1, (short)0, acc, false, false);
    acc = __builtin_amdgcn_wmma_f32_16x16x4_f32(false, a2, false, bw1_2, (short)0, acc, false, false);

    // ReLU + D-layout -> A-layout transpose via LDS (same-wave, in-order DS pipe)
#pragma unroll
    for (int r = 0; r < 8; ++r) {
      hb[(r + 8 * hi) * 16 + n] = __builtin_fmaxf(acc[r], 0.f);
    }
    v2f h0 = *(const v2f*)&hb[n * 16 + 0 + 2 * hi];
    v2f h1 = *(const v2f*)&hb[n * 16 + 4 + 2 * hi];
    v2f h2 = *(const v2f*)&hb[n * 16 + 8 + 2 * hi];

    // GEMM2: msg = h @ W2 + b2
    v8f acc2 = {c2, c2, c2, c2, c2, c2, c2, c2};
    acc2 = __builtin_amdgcn_wmma_f32_16x16x4_f32(false, h0, false, bw2_0, (short)0, acc2, false, false);
    acc2 = __builtin_amdgcn_wmma_f32_16x16x4_f32(false, h1, false, bw2_1, (short)0, acc2, false, false);
    acc2 = __builtin_amdgcn_wmma_f32_16x16x4_f32(false, h2, false, bw2_2, (short)0, acc2, false, false);

    // Segment-sum scatter: lane owns feature n of rows (r + 8*hi).
    // 32-bit element offsets (max ~1.2M) -> SGPR-base + voffset atomics.
    if (n < 12) {
      const int* rbase = receivers + e0 + 8 * hi;
#pragma unroll
      for (int r = 0; r < 8; ++r) {
        int off = rbase[r] * 12 + n;
        unsafeAtomicAdd(out + off, acc2[r]);
      }
    }
  }
}

__global__ void tail_kernel(const float* __restrict__ edges,
                            const float* __restrict__ bn_scale,
                            const float* __restrict__ bn_bias,
                            const float* __restrict__ bn_mean,
                            const float* __restrict__ bn_var,
                            const float* __restrict__ W1,
                            const float* __restrict__ b1,
                            const float* __restrict__ W2,
                            const float* __restrict__ b2,
                            const int* __restrict__ receivers,
                            float* __restrict__ out, long ebase, int count) {
  int t = blockIdx.x * blockDim.x + threadIdx.x;
  if (t >= count) return;
  long e = ebase + t;
  float x[12], h[12];
#pragma unroll
  for (int k = 0; k < 12; ++k) {
    float s = bn_scale[k] * rsqrtf(bn_var[k] + BN_EPS);
    x[k] = (edges[e * 12 + k] - bn_mean[k]) * s + bn_bias[k];
  }
#pragma unroll
  for (int j = 0; j < 12; ++j) {
    float a = b1[j];
#pragma unroll
    for (int k = 0; k < 12; ++k) a += x[k] * W1[k * 12 + j];
    h[j] = a > 0.f ? a : 0.f;
  }
  int rcv = receivers[e];
#pragma unroll
  for (int j = 0; j < 12; ++j) {
    float a = b2[j];
#pragma unroll
    for (int k = 0; k < 12; ++k) a += h[k] * W2[k * 12 + j];
    unsafeAtomicAdd(out + (long)rcv * 12 + j, a);
  }
}

extern "C" void kernel_launch(void* const* d_in, const int* in_sizes, int n_in,
                              void* d_out, int out_size, void* d_ws, size_t ws_size,
                              hipStream_t stream) {
  const float* edges    = (const float*)d_in[0];
  const float* bn_scale = (const float*)d_in[1];
  const float* bn_bias  = (const float*)d_in[2];
  const float* bn_mean  = (const float*)d_in[3];
  const float* bn_var   = (const float*)d_in[4];
  const float* W1       = (const float*)d_in[5];
  const float* b1       = (const float*)d_in[6];
  const float* W2       = (const float*)d_in[7];
  const float* b2       = (const float*)d_in[8];
  const int* receivers  = (const int*)d_in[9];
  float* out = (float*)d_out;
  float* ws  = (float*)d_ws;

  long E = (long)in_sizes[0] / 12;
  long ntiles = E / 16;
  long rem = E % 16;

  zero_kernel<<<(out_size + 255) / 256, 256, 0, stream>>>(out, out_size);
  prep_kernel<<<1, 128, 0, stream>>>(bn_scale, bn_bias, bn_mean, bn_var,
                                     W1, b1, W2, b2, ws);
  if (ntiles > 0) {
    // Persistent-ish grid-stride: ~8 tiles per wave amortizes constant loads.
    long blocks = (ntiles + 63) / 64;
    if (blocks > 16384) blocks = 16384;
    edge_wmma_kernel<<<(int)blocks, 256, 0, stream>>>(edges, receivers, ws, out,
                                                      (int)ntiles);
  }
  if (rem > 0) {
    tail_kernel<<<1, 32, 0, stream>>>(edges, bn_scale, bn_bias, bn_mean, bn_var,
                                      W1, b1, W2, b2, receivers, out,
                                      ntiles * 16, (int)rem);
  }
}